// sub_model_84756884619864
// MI455X (gfx1250) — compile-verified
//
#include <hip/hip_runtime.h>

// Problem constants (from reference)
#define BB      4
#define DSTATE  16
#define KDIRS   4
#define DINNER  192
#define RTRANK  6
#define LL      4096      // H*W (64x64)
#define NPROJ   38        // R + 2N
#define RSTRIDE 40        // padded row stride for Rt (float4 friendly)
#define CHUNKS  16        // chunked parallel scan
#define CLEN    (LL / CHUNKS)   // 256 steps per chunk

typedef __attribute__((ext_vector_type(2))) float v2f;
typedef __attribute__((ext_vector_type(8))) float v8f;

// ---------------------------------------------------------------------------
// Kernel 1: transpose x[b, d, s] -> xt[b, s, d]  (scan lanes read u coalesced)
// ---------------------------------------------------------------------------
__global__ void transpose_xt(const float* __restrict__ x, float* __restrict__ xt) {
  __shared__ float tile[32][33];
  const int b  = blockIdx.z;
  const int s0 = blockIdx.x * 32;
  const int d0 = blockIdx.y * 32;
  const int tx = threadIdx.x;      // 0..31
  const int ty = threadIdx.y;      // 0..7
  const float* xb = x + (size_t)b * DINNER * LL;
  #pragma unroll
  for (int i = 0; i < 32; i += 8)
    tile[ty + i][tx] = xb[(size_t)(d0 + ty + i) * LL + (s0 + tx)];
  __syncthreads();
  float* xtb = xt + (size_t)b * LL * DINNER;
  #pragma unroll
  for (int i = 0; i < 32; i += 8)
    xtb[(size_t)(s0 + ty + i) * DINNER + (d0 + tx)] = tile[tx][ty + i];
}

// ---------------------------------------------------------------------------
// Kernel 2: projection GEMM with fp32 WMMA (V_WMMA_F32_16X16X4_F32).
//   Per (b,k):  R = Wproj[k] (38x192) @ X0[b] (192x4096), all row-major.
//   All 4 scan directions are column-bijections of this product, so no gathers.
//   LDS holds weights as float2 K-pairs: A fragment = one ds_load_b64 straight
//   into an even-aligned VGPR pair (no mov fixups, conflict-free across lanes).
// ---------------------------------------------------------------------------
__global__ void proj_gemm(const float* __restrict__ x,
                          const float* __restrict__ wproj,
                          float* __restrict__ Rt) {
  __shared__ v2f Wlds2[(DINNER / 2) * 48];   // [kpair][row], 36.9 KB
  const int bk = blockIdx.y;                 // b*4 + k
  const int b  = bk >> 2;
  const int k  = bk & 3;
  const int tid = threadIdx.x;

  const float* Wk = wproj + (size_t)k * NPROJ * DINNER;
  for (int i = tid; i < (DINNER / 2) * 48; i += 256) {
    const int kp  = i / 48;          // K pair index (covers kk=2*kp, 2*kp+1)
    const int row = i - kp * 48;
    v2f wv;
    wv.x = (row < NPROJ) ? Wk[(size_t)row * DINNER + 2 * kp]     : 0.0f;
    wv.y = (row < NPROJ) ? Wk[(size_t)row * DINNER + 2 * kp + 1] : 0.0f;
    Wlds2[i] = wv;
  }
  __syncthreads();

  const int wave = tid >> 5;
  const int lane = tid & 31;
  const int hi   = lane >> 4;           // 0: lanes 0-15, 1: lanes 16-31
  const int ln   = lane & 15;
  const int l0   = blockIdx.x * 128 + wave * 16;

  const float* xb = x + (size_t)b * DINNER * LL;

  v8f acc0 = {}; v8f acc1 = {}; v8f acc2 = {};

  for (int kk = 0; kk < DINNER; kk += 4) {
    const int kA  = kk + 2 * hi;          // K rows (kA, kA+1) for this half
    const int kp  = (kk >> 1) + hi;       // pair index
    // B fragment: X rows kA, kA+1 at column l0+ln (coalesced 16-float runs)
    v2f bf;
    bf.x = xb[(size_t)kA * LL + (l0 + ln)];
    bf.y = xb[(size_t)(kA + 1) * LL + (l0 + ln)];
    // A fragments: single b64 LDS loads
    const v2f a0 = Wlds2[kp * 48 + ln];
    const v2f a1 = Wlds2[kp * 48 + 16 + ln];
    const v2f a2 = Wlds2[kp * 48 + 32 + ln];

    acc0 = __builtin_amdgcn_wmma_f32_16x16x4_f32(false, a0, false, bf, (short)0, acc0, false, false);
    acc1 = __builtin_amdgcn_wmma_f32_16x16x4_f32(false, a1, false, bf, (short)0, acc1, false, false);
    acc2 = __builtin_amdgcn_wmma_f32_16x16x4_f32(false, a2, false, bf, (short)0, acc2, false, false);
  }

  // Store transposed: lane ln = column (l), vgpr v + 8*hi = row (r)
  float* rowp = Rt + ((size_t)bk * LL + (l0 + ln)) * RSTRIDE;
  #pragma unroll
  for (int v = 0; v < 8; ++v) {
    const int r0 = v + 8 * hi;
    rowp[r0]      = acc0[v];                 // rows 0..15
    rowp[16 + r0] = acc1[v];                 // rows 16..31
    const int r2 = 32 + r0;                  // rows 32..47 (keep < 38)
    if (r2 < NPROJ) rowp[r2] = acc2[v];
  }
}

// ---------------------------------------------------------------------------
// Kernel 3: zero the output (merge accumulates with atomics)
// ---------------------------------------------------------------------------
__global__ void zero_out(float* __restrict__ out, int n) {
  const int i = blockIdx.x * 256 + threadIdx.x;
  if (i < n) out[i] = 0.0f;
}

// ---------------------------------------------------------------------------
// Chunk-parallel selective scan.
//   h <- exp(dt*A)*h + (dt*u)*B  is a diagonal linear recurrence; the decay
//   over a chunk is exp(A[n] * sum(dt)), so a chunk exports only a scalar
//   dt-sum and h_end[16].  16 chunks x 16 (b,k) x 192 channels -> 1536 waves
//   (vs 96 for a monolithic scan).  Pass1/Pass3 share one templated body so
//   dt is computed bitwise-identically (deterministic).
// ---------------------------------------------------------------------------
template <bool FIRST>
__global__ void scan_chunk(const float* __restrict__ xt,
                           const float* __restrict__ Rt,
                           const float* __restrict__ dtW,
                           const float* __restrict__ dtB,
                           const float* __restrict__ Alogs,
                           const float* __restrict__ Ds,
                           const float* __restrict__ Hstart,
                           float* __restrict__ hend,
                           float* __restrict__ Ssum,
                           float* __restrict__ out) {
  const int c  = blockIdx.x;         // chunk
  const int bk = blockIdx.y;         // b*4 + k
  const int b  = bk >> 2;
  const int k  = bk & 3;
  const int d  = threadIdx.x;        // 0..191
  const int kd = k * DINNER + d;

  float A[DSTATE];
  #pragma unroll
  for (int n = 0; n < DSTATE; ++n) A[n] = -__expf(Alogs[(size_t)kd * DSTATE + n]);

  float w[RTRANK];
  #pragma unroll
  for (int r = 0; r < RTRANK; ++r) w[r] = dtW[(size_t)kd * RTRANK + r];

  const float bias   = dtB[kd];
  const float dscale = Ds[kd];

  const size_t cbase = (((size_t)bk * CHUNKS + c) * DINNER + d) * DSTATE;

  float h[DSTATE];
  if (FIRST) {
    #pragma unroll
    for (int n = 0; n < DSTATE; ++n) h[n] = 0.0f;
  } else {
    #pragma unroll
    for (int n = 0; n < DSTATE; ++n) h[n] = Hstart[cbase + n];
  }

  const float* rtb = Rt + (size_t)bk * LL * RSTRIDE;
  const float* xtb = xt + (size_t)b * LL * DINNER;
  float* outc = out + (size_t)(b * DINNER + d) * LL;

  float S = 0.0f;

  for (int i = 0; i < CLEN; ++i) {
    const int l  = c * CLEN + i;
    // tau_k(l): traversal order -> spatial index
    const int lf = (k >= 2) ? (LL - 1 - l) : l;
    const int lu = (k & 1) ? (((lf & 63) << 6) | (lf >> 6)) : lf;

    // prefetch next step's projected row (gfx1250 global_prefetch)
    if (i + 1 < CLEN) {
      const int lf2 = (k >= 2) ? (lf - 1) : (lf + 1);
      const int lu2 = (k & 1) ? (((lf2 & 63) << 6) | (lf2 >> 6)) : lf2;
      __builtin_prefetch(rtb + (size_t)lu2 * RSTRIDE, 0, 3);
    }

    // wave-uniform broadcast load of 38 projected values (10x b128)
    const float4* row = (const float4*)(rtb + (size_t)lu * RSTRIDE);
    const float4 q0 = row[0], q1 = row[1], q2 = row[2], q3 = row[3], q4 = row[4];
    const float4 q5 = row[5], q6 = row[6], q7 = row[7], q8 = row[8], q9 = row[9];

    // dt = softplus(<row[0..5], w> + bias)
    float dtr = bias;
    dtr = fmaf(q0.x, w[0], dtr); dtr = fmaf(q0.y, w[1], dtr);
    dtr = fmaf(q0.z, w[2], dtr); dtr = fmaf(q0.w, w[3], dtr);
    dtr = fmaf(q1.x, w[4], dtr); dtr = fmaf(q1.y, w[5], dtr);
    const float dt = (dtr > 20.0f) ? dtr : __logf(1.0f + __expf(dtr));
    if (FIRST) S += dt;

    const float u  = xtb[(size_t)lu * DINNER + d];   // coalesced across lanes
    const float du = dt * u;

    const float Bv[DSTATE] = { q1.z, q1.w, q2.x, q2.y, q2.z, q2.w, q3.x, q3.y,
                               q3.z, q3.w, q4.x, q4.y, q4.z, q4.w, q5.x, q5.y };
    const float Cv[DSTATE] = { q5.z, q5.w, q6.x, q6.y, q6.z, q6.w, q7.x, q7.y,
                               q7.z, q7.w, q8.x, q8.y, q8.z, q8.w, q9.x, q9.y };

    float y = 0.0f;
    #pragma unroll
    for (int n = 0; n < DSTATE; ++n) {
      const float dA = __expf(dt * A[n]);
      h[n] = fmaf(h[n], dA, du * Bv[n]);
      if (!FIRST) y = fmaf(h[n], Cv[n], y);
    }

    if (!FIRST) unsafeAtomicAdd(&outc[lu], y + dscale * u);  // CrossMerge
  }

  if (FIRST) {
    #pragma unroll
    for (int n = 0; n < DSTATE; ++n) hend[cbase + n] = h[n];
    Ssum[((size_t)bk * CHUNKS + c) * DINNER + d] = S;
  }
}

// ---------------------------------------------------------------------------
// Pass 2: serial carry across the 16 chunks (per channel, trivial work).
//   H_c entry state: H <- exp(A * S_c) * H + h_end_c
// ---------------------------------------------------------------------------
__global__ void scan_combine(const float* __restrict__ Alogs,
                             const float* __restrict__ hend,
                             const float* __restrict__ Ssum,
                             float* __restrict__ Hstart) {
  const int bk = blockIdx.x;
  const int k  = bk & 3;
  const int d  = threadIdx.x;
  const int kd = k * DINNER + d;

  float A[DSTATE];
  #pragma unroll
  for (int n = 0; n < DSTATE; ++n) A[n] = -__expf(Alogs[(size_t)kd * DSTATE + n]);

  float H[DSTATE];
  #pragma unroll
  for (int n = 0; n < DSTATE; ++n) H[n] = 0.0f;

  for (int c = 0; c < CHUNKS; ++c) {
    const size_t cbase = (((size_t)bk * CHUNKS + c) * DINNER + d) * DSTATE;
    #pragma unroll
    for (int n = 0; n < DSTATE; ++n) Hstart[cbase + n] = H[n];
    const float S = Ssum[((size_t)bk * CHUNKS + c) * DINNER + d];
    #pragma unroll
    for (int n = 0; n < DSTATE; ++n)
      H[n] = fmaf(__expf(A[n] * S), H[n], hend[cbase + n]);
  }
}

// ---------------------------------------------------------------------------
extern "C" void kernel_launch(void* const* d_in, const int* in_sizes, int n_in,
                              void* d_out, int out_size, void* d_ws, size_t ws_size,
                              hipStream_t stream) {
  const float* x      = (const float*)d_in[0];  // [B, D, H, W]
  const float* wproj  = (const float*)d_in[1];  // [K, 38, 192]
  const float* dtw    = (const float*)d_in[2];  // [K, 192, 6]
  const float* dtb    = (const float*)d_in[3];  // [K, 192]
  const float* alogs  = (const float*)d_in[4];  // [K*192, 16]
  const float* ds     = (const float*)d_in[5];  // [K*192]
  float* out = (float*)d_out;                   // [B, D, H, W]

  float* xt     = (float*)d_ws;                                   // 12.58 MB
  float* Rt     = xt     + (size_t)BB * LL * DINNER;              // 10.49 MB
  float* hend   = Rt     + (size_t)BB * KDIRS * LL * RSTRIDE;     //  3.15 MB
  float* Hstart = hend   + (size_t)BB * KDIRS * CHUNKS * DINNER * DSTATE; // 3.15 MB
  float* Ssum   = Hstart + (size_t)BB * KDIRS * CHUNKS * DINNER * DSTATE; // 0.20 MB
  (void)in_sizes; (void)n_in; (void)ws_size;

  transpose_xt<<<dim3(LL / 32, DINNER / 32, BB), dim3(32, 8), 0, stream>>>(x, xt);
  proj_gemm<<<dim3(LL / 128, BB * KDIRS), 256, 0, stream>>>(x, wproj, Rt);
  zero_out<<<(out_size + 255) / 256, 256, 0, stream>>>(out, out_size);

  scan_chunk<true><<<dim3(CHUNKS, BB * KDIRS), DINNER, 0, stream>>>(
      xt, Rt, dtw, dtb, alogs, ds, nullptr, hend, Ssum, out);
  scan_combine<<<BB * KDIRS, DINNER, 0, stream>>>(alogs, hend, Ssum, Hstart);
  scan_chunk<false><<<dim3(CHUNKS, BB * KDIRS), DINNER, 0, stream>>>(
      xt, Rt, dtw, dtb, alogs, ds, Hstart, hend, Ssum, out);
}